// RRN_17746804867120
// MI455X (gfx1250) — compile-verified
//
#include <hip/hip_runtime.h>

#define HDIM 96
#define NCELL 81
#define NEDGE 20
#define LSTRIDE 100   // LDS row stride in floats: 16B-aligned rows, conflict-free b64 reads

typedef __attribute__((ext_vector_type(2))) float v2f;
typedef __attribute__((ext_vector_type(8))) float v8f;

static __device__ __forceinline__ v8f wmma4(v2f a, v2f b, v8f c) {
  // D = A(16x4 f32) * B(4x16 f32) + C(16x16 f32)
  return __builtin_amdgcn_wmma_f32_16x16x4_f32(false, a, false, b, (short)0, c,
                                               false, false);
}

// out = act( A(Mrows x K) @ W(K x Nout, row-major) + bscale*bias + addend )
// Block: 128 threads = 4 waves; tile = 64 rows x 32 cols (2 N-subtiles/wave,
// A-fragment reused for both). K in {16, 96}.
__global__ __launch_bounds__(128) void gemm_wmma(
    const float* __restrict__ A, const float* __restrict__ W,
    const float* __restrict__ bias, const float* __restrict__ addend,
    float* __restrict__ out, int Mrows, int K, int Nout, int do_relu,
    float bscale)
{
  __shared__ float As[64 * LSTRIDE];
  __shared__ float Ws[32 * LSTRIDE];
  const int tid = threadIdx.x;
  const int mb0 = blockIdx.x * 64;
  const int nb0 = blockIdx.y * 32;
  const int K4 = K >> 2;

  // Stage A tile (zero-fill rows past Mrows)
  for (int t = tid; t < 64 * K4; t += 128) {
    int i = t / K4;
    int k = (t - i * K4) << 2;
    int m = mb0 + i;
    float4 v = make_float4(0.f, 0.f, 0.f, 0.f);
    if (m < Mrows) v = *(const float4*)(A + (size_t)m * K + k);
    *(float4*)(&As[i * LSTRIDE + k]) = v;
  }
  // Stage W tile transposed: Ws[n][k] (zero-fill cols past Nout)
  for (int t = tid; t < 32 * K; t += 128) {
    int n = t / K;
    int k = t - n * K;
    int gn = nb0 + n;
    Ws[n * LSTRIDE + k] = (gn < Nout) ? W[(size_t)k * Nout + gn] : 0.f;
  }
  __syncthreads();

  const int wave = tid >> 5;
  const int lane = tid & 31;
  const int l16 = lane & 15;
  const int half = lane >> 4;
  const float* Ap  = &As[(wave * 16 + l16) * LSTRIDE];
  const float* Bp0 = &Ws[l16 * LSTRIDE];
  const float* Bp1 = &Ws[(16 + l16) * LSTRIDE];

  v8f acc0 = {0.f, 0.f, 0.f, 0.f, 0.f, 0.f, 0.f, 0.f};
  v8f acc1 = {0.f, 0.f, 0.f, 0.f, 0.f, 0.f, 0.f, 0.f};
  for (int kk = 0; kk < K; kk += 4) {
    const int ko = kk + half * 2;
    v2f a;  a.x  = Ap[ko];  a.y  = Ap[ko + 1];
    v2f b0; b0.x = Bp0[ko]; b0.y = Bp0[ko + 1];
    v2f b1; b1.x = Bp1[ko]; b1.y = Bp1[ko + 1];
    acc0 = wmma4(a, b0, acc0);
    acc1 = wmma4(a, b1, acc1);
  }

  const int gn0 = nb0 + l16;
  const int gn1 = nb0 + 16 + l16;
  float bv0 = 0.f, bv1 = 0.f;
  if (bias != nullptr) {
    if (gn0 < Nout) bv0 = bscale * bias[gn0];
    if (gn1 < Nout) bv1 = bscale * bias[gn1];
  }
#pragma unroll
  for (int v = 0; v < 8; ++v) {
    int m = mb0 + wave * 16 + v + half * 8;
    if (m < Mrows) {
      if (gn0 < Nout) {
        float val = acc0[v] + bv0;
        if (addend != nullptr) val += addend[(size_t)m * Nout + gn0];
        if (do_relu) val = fmaxf(val, 0.f);
        out[(size_t)m * Nout + gn0] = val;
      }
      if (gn1 < Nout) {
        float val = acc1[v] + bv1;
        if (addend != nullptr) val += addend[(size_t)m * Nout + gn1];
        if (do_relu) val = fmaxf(val, 0.f);
        out[(size_t)m * Nout + gn1] = val;
      }
    }
  }
}

// Fused edge message kernel. Per block: 16 cells; loop over 20 neighbors:
//   A = relu(P[cell] + Q[qbase+edge])            (16 x 96, in LDS)
//   S += relu(A @ W1 + b1)                       (accumulated in registers)
// Block: 96 threads = 3 waves; wave w owns output columns [32w, 32w+32)
// (2 accumulators per wave, A-fragment reused). P tile staged in LDS once;
// next neighbor's Q rows prefetched (global_prefetch_b8).
__global__ __launch_bounds__(96) void edge_msg(
    const float* __restrict__ P, const float* __restrict__ Q,
    const int* __restrict__ edges, const float* __restrict__ W1,
    const float* __restrict__ b1, float* __restrict__ S, int Mrows)
{
  __shared__ float Wl[HDIM * LSTRIDE];  // W1 transposed: Wl[n][k]
  __shared__ float Pl[16 * LSTRIDE];
  __shared__ float Al[16 * LSTRIDE];
  __shared__ int el[16 * NEDGE];
  __shared__ int qb[16];
  const int tid = threadIdx.x;
  const int r0 = blockIdx.x * 16;

  for (int t = tid; t < HDIM * HDIM; t += 96) {
    int n = t / HDIM;
    int k = t - n * HDIM;
    Wl[n * LSTRIDE + k] = W1[(size_t)k * HDIM + n];
  }
  for (int t = tid; t < 16 * (HDIM / 4); t += 96) {   // P tile, float4
    int i = t / 24;
    int k = (t - i * 24) << 2;
    int r = r0 + i; if (r >= Mrows) r = 0;
    *(float4*)(&Pl[i * LSTRIDE + k]) =
        *(const float4*)(P + (size_t)r * HDIM + k);
  }
  for (int t = tid; t < 16 * NEDGE; t += 96) {
    int i = t / NEDGE;
    int r = r0 + i; if (r >= Mrows) r = 0;
    el[t] = edges[(r % NCELL) * NEDGE + (t - i * NEDGE)];
  }
  if (tid < 16) {
    int r = r0 + tid; if (r >= Mrows) r = 0;
    qb[tid] = (r / NCELL) * NCELL;
  }

  const int wave = tid >> 5;
  const int lane = tid & 31;
  const int l16 = lane & 15;
  const int half = lane >> 4;
  const int gn0 = wave * 32 + l16;         // global output columns
  const int gn1 = wave * 32 + 16 + l16;
  const float b1v0 = b1[gn0];
  const float b1v1 = b1[gn1];
  const float* Ap  = &Al[l16 * LSTRIDE];
  const float* Bp0 = &Wl[gn0 * LSTRIDE];
  const float* Bp1 = &Wl[gn1 * LSTRIDE];

  float sacc0[8], sacc1[8];
#pragma unroll
  for (int v = 0; v < 8; ++v) { sacc0[v] = 0.f; sacc1[v] = 0.f; }

  for (int j = 0; j < NEDGE; ++j) {
    __syncthreads();  // covers init loads (j==0) and prev-round A reads
    for (int t = tid; t < 16 * (HDIM / 4); t += 96) {
      int i = t / 24;
      int k = (t - i * 24) << 2;
      int qr = qb[i] + el[i * NEDGE + j];
      float4 q = *(const float4*)(Q + (size_t)qr * HDIM + k);
      float4 p = *(const float4*)(&Pl[i * LSTRIDE + k]);
      float4 o;
      o.x = fmaxf(p.x + q.x, 0.f);
      o.y = fmaxf(p.y + q.y, 0.f);
      o.z = fmaxf(p.z + q.z, 0.f);
      o.w = fmaxf(p.w + q.w, 0.f);
      *(float4*)(&Al[i * LSTRIDE + k]) = o;
      if (k == 0 && j + 1 < NEDGE) {    // prefetch next neighbor's row
        int qr2 = qb[i] + el[i * NEDGE + j + 1];
        __builtin_prefetch(Q + (size_t)qr2 * HDIM, 0, 1);
      }
    }
    __syncthreads();
    v8f d0 = {0.f, 0.f, 0.f, 0.f, 0.f, 0.f, 0.f, 0.f};
    v8f d1 = {0.f, 0.f, 0.f, 0.f, 0.f, 0.f, 0.f, 0.f};
    for (int kk = 0; kk < HDIM; kk += 4) {
      const int ko = kk + half * 2;
      v2f a;  a.x  = Ap[ko];  a.y  = Ap[ko + 1];
      v2f b0; b0.x = Bp0[ko]; b0.y = Bp0[ko + 1];
      v2f b1f; b1f.x = Bp1[ko]; b1f.y = Bp1[ko + 1];
      d0 = wmma4(a, b0, d0);
      d1 = wmma4(a, b1f, d1);
    }
#pragma unroll
    for (int v = 0; v < 8; ++v) {
      sacc0[v] += fmaxf(d0[v] + b1v0, 0.f);
      sacc1[v] += fmaxf(d1[v] + b1v1, 0.f);
    }
  }

#pragma unroll
  for (int v = 0; v < 8; ++v) {
    int r = r0 + v + half * 8;
    if (r < Mrows) {
      S[(size_t)r * HDIM + gn0] = sacc0[v];
      S[(size_t)r * HDIM + gn1] = sacc1[v];
    }
  }
}

__global__ void lstm_update(const float* __restrict__ gates,
                            float* __restrict__ c, float* __restrict__ h,
                            int total)
{
  int idx = blockIdx.x * blockDim.x + threadIdx.x;
  if (idx >= total) return;
  int row = idx / HDIM;
  int col = idx - row * HDIM;
  const float* g = gates + (size_t)row * 384;
  float ig = g[col], fg = g[col + 96], gg = g[col + 192], og = g[col + 288];
  float si = 1.f / (1.f + __expf(-ig));
  float sf = 1.f / (1.f + __expf(-fg));
  float so = 1.f / (1.f + __expf(-og));
  float cn = sf * c[idx] + si * tanhf(gg);
  c[idx] = cn;
  h[idx] = so * tanhf(cn);
}

__global__ void pack_weights(const float* __restrict__ w_ih,
                             const float* __restrict__ w_hh,
                             const float* __restrict__ b_ih,
                             const float* __restrict__ b_hh,
                             float* __restrict__ wihT, float* __restrict__ whhT,
                             float* __restrict__ bg)
{
  int idx = blockIdx.x * blockDim.x + threadIdx.x;
  if (idx < 384 * 96) {
    int n = idx / 96;
    int k = idx - n * 96;
    wihT[k * 384 + n] = w_ih[idx];
    whhT[k * 384 + n] = w_hh[idx];
  }
  if (idx < 384) bg[idx] = b_ih[idx] + b_hh[idx];
}

__global__ void gather_embed(const int* __restrict__ grids,
                             const float* __restrict__ embed,
                             float* __restrict__ X0, int R)
{
  int idx = blockIdx.x * blockDim.x + threadIdx.x;
  if (idx >= R * 16) return;
  int row = idx >> 4;
  int k = idx & 15;
  X0[idx] = embed[grids[row] * 16 + k];
}

extern "C" void kernel_launch(void* const* d_in, const int* in_sizes, int n_in,
                              void* d_out, int out_size, void* d_ws,
                              size_t ws_size, hipStream_t stream)
{
  const int*   grids = (const int*)d_in[0];
  const int*   edges = (const int*)d_in[1];
  const float* embed = (const float*)d_in[2];
  const float *iW0 = (const float*)d_in[3],  *ib0 = (const float*)d_in[4];
  const float *iW1 = (const float*)d_in[5],  *ib1 = (const float*)d_in[6];
  const float *iW2 = (const float*)d_in[7],  *ib2 = (const float*)d_in[8];
  const float *rW0 = (const float*)d_in[9],  *rb0 = (const float*)d_in[10];
  const float *rW1 = (const float*)d_in[11], *rb1 = (const float*)d_in[12];
  const float *rW2 = (const float*)d_in[13], *rb2 = (const float*)d_in[14];
  const float *gW0 = (const float*)d_in[15], *gb0 = (const float*)d_in[16];
  const float *gW1 = (const float*)d_in[17], *gb1 = (const float*)d_in[18];
  const float *gW2 = (const float*)d_in[19], *gb2 = (const float*)d_in[20];
  const float *w_ih = (const float*)d_in[21];
  const float *w_hh = (const float*)d_in[22];
  const float *b_ih = (const float*)d_in[23];
  const float *b_hh = (const float*)d_in[24];
  const float *oW0 = (const float*)d_in[25], *ob0 = (const float*)d_in[26];
  const float *oW1 = (const float*)d_in[27], *ob1 = (const float*)d_in[28];
  const float *oW2 = (const float*)d_in[29], *ob2 = (const float*)d_in[30];
  const float *c0  = (const float*)d_in[31];

  const int R = in_sizes[0];                // B * 81 = 10368
  int iters = out_size / (R * 9);
  if (iters < 1) iters = 1;

  float* ws = (float*)d_ws;
  size_t off = 0;
  auto alloc = [&](size_t n) -> float* { float* p = ws + off; off += n; return p; };
  float* X0    = alloc((size_t)R * 16);
  float* X     = alloc((size_t)R * 96);
  float* XG    = alloc((size_t)R * 96);
  float* hbuf  = alloc((size_t)R * 96);
  float* cbuf  = alloc((size_t)R * 96);
  float* P     = alloc((size_t)R * 96);
  float* Q     = alloc((size_t)R * 96);
  float* S     = alloc((size_t)R * 96);
  float* Msg   = alloc((size_t)R * 96);
  float* t1    = alloc((size_t)R * 96);     // scratch (g_mlp/r_mlp hidden)
  float* t2    = alloc((size_t)R * 96);
  float* inp   = alloc((size_t)R * 96);
  float* gates = alloc((size_t)R * 384);
  float* wihT  = alloc((size_t)96 * 384);
  float* whhT  = alloc((size_t)96 * 384);
  float* bg    = alloc(384);

  auto gemm = [&](const float* A, const float* W, const float* bias,
                  const float* add, float* out_, int Mr, int K, int N,
                  int relu, float bs) {
    dim3 grid((Mr + 63) / 64, (N + 31) / 32);
    gemm_wmma<<<grid, dim3(128), 0, stream>>>(A, W, bias, add, out_, Mr, K, N,
                                              relu, bs);
  };

  // ---- iteration-invariant prepass ----
  pack_weights<<<(384 * 96 + 255) / 256, 256, 0, stream>>>(w_ih, w_hh, b_ih,
                                                           b_hh, wihT, whhT, bg);
  gather_embed<<<(R * 16 + 255) / 256, 256, 0, stream>>>(grids, embed, X0, R);
  gemm(X0, iW0, ib0, nullptr, t1, R, 16, 96, 1, 1.f);      // input_mlp L1
  gemm(t1, iW1, ib1, nullptr, t2, R, 96, 96, 1, 1.f);      // input_mlp L2
  gemm(t2, iW2, ib2, nullptr, X,  R, 96, 96, 0, 1.f);      // X (linear)
  gemm(X, gW0, gb0, nullptr, XG, R, 96, 96, 0, 1.f);       // X @ G0a + gb0
  hipMemcpyAsync(hbuf, X,  (size_t)R * 96 * sizeof(float),
                 hipMemcpyDeviceToDevice, stream);
  hipMemcpyAsync(cbuf, c0, (size_t)R * 96 * sizeof(float),
                 hipMemcpyDeviceToDevice, stream);

  // ---- recurrent iterations ----
  for (int it = 0; it < iters; ++it) {
    gemm(hbuf, rW0,           rb0,     nullptr, P, R, 96, 96, 0, 1.f);  // src half + b0
    gemm(hbuf, rW0 + 96 * 96, nullptr, nullptr, Q, R, 96, 96, 0, 1.f);  // nbr half
    edge_msg<<<(R + 15) / 16, dim3(96), 0, stream>>>(P, Q, edges, rW1, rb1,
                                                     S, R);
    gemm(S, rW2, rb2, nullptr, Msg, R, 96, 96, 0, 20.f);   // M = S@W2 + 20*b2
    gemm(Msg, gW0 + 96 * 96, nullptr, XG, t1, R, 96, 96, 1, 1.f);  // g_mlp L1
    gemm(t1, gW1, gb1, nullptr, t2,  R, 96, 96, 1, 1.f);           // g_mlp L2
    gemm(t2, gW2, gb2, nullptr, inp, R, 96, 96, 0, 1.f);           // g_mlp L3
    gemm(inp,  wihT, bg,      nullptr, gates, R, 96, 384, 0, 1.f); // inp@Wih^T
    gemm(hbuf, whhT, nullptr, gates,   gates, R, 96, 384, 0, 1.f); // + h@Whh^T
    lstm_update<<<(R * 96 + 255) / 256, 256, 0, stream>>>(gates, cbuf, hbuf,
                                                          R * 96);
    gemm(hbuf, oW0, ob0, nullptr, t1, R, 96, 96, 1, 1.f);          // r_mlp L1
    gemm(t1,   oW1, ob1, nullptr, t2, R, 96, 96, 1, 1.f);          // r_mlp L2
    gemm(t2,   oW2, ob2, nullptr, (float*)d_out + (size_t)it * R * 9,
         R, 96, 9, 0, 1.f);                                        // logits
  }
}